// MessagePassingLayer_25400436589080
// MI455X (gfx1250) — compile-verified
//
#include <hip/hip_runtime.h>
#include <math.h>

#define N_NODES 100000
#define N_EDGES 1250000
#define D 64
#define NT (N_NODES / 16)   // 6250 tiles of 16 rows (exact)
#define KS1 66              // K-major LDS stride for K=64 weights (even -> 8B-aligned pairs)
#define KS2 130             // K-major LDS stride for K=128 weights
#define HSTR 68             // padded LDS stride for activation tiles
#define LN_EPS 1e-3f

typedef float v2f __attribute__((ext_vector_type(2)));
typedef float v8f __attribute__((ext_vector_type(8)));

// D = A(16x4,f32) * B(4x16,f32) + C(16x16,f32)  -> v_wmma_f32_16x16x4_f32
__device__ __forceinline__ v8f wmma4(v2f a, v2f b, v8f c) {
    return __builtin_amdgcn_wmma_f32_16x16x4_f32(
        /*neg_a=*/false, a, /*neg_b=*/false, b,
        /*c_mod=*/(short)0, c, /*reuse_a=*/false, /*reuse_b=*/false);
}

// Contiguous 8-byte B-fragment load from K-major LDS weights: elements (k, k+1) of column n.
__device__ __forceinline__ v2f ldsB(const float* wt, int stride, int col, int k) {
    return *(const v2f*)(wt + col * stride + k);   // 8B aligned: col*stride even, k even
}

// ---------------------------------------------------------------------------
// Kernel 1: messages = relu(neighbor @ W1 + b1) @ W2 + b2
// One wave per 16-row tile; 4 waves per block. Weights K-major in LDS.
// ---------------------------------------------------------------------------
__global__ __launch_bounds__(128) void msg_mlp_kernel(
    const float* __restrict__ X,
    const float* __restrict__ W1, const float* __restrict__ b1,
    const float* __restrict__ W2, const float* __restrict__ b2,
    float* __restrict__ Msg)
{
    __shared__ float w1t[D * KS1];       // [n][k] transposed
    __shared__ float w2t[D * KS1];
    __shared__ float hs[4][16 * HSTR];

    const int tid = threadIdx.x;
    for (int i = tid; i < D * D; i += 128) {
        int k = i >> 6, n = i & 63;      // W is [k][n] row-major in global
        w1t[n * KS1 + k] = W1[i];
        w2t[n * KS1 + k] = W2[i];
    }
    __syncthreads();

    const int wid  = tid >> 5;
    const int lane = tid & 31;
    const int half = lane >> 4;   // K half-select (ISA 16x4 A layout)
    const int mrow = lane & 15;   // M row within tile
    const int tile = blockIdx.x * 4 + wid;
    const bool active = tile < NT;
    const int m0 = tile * 16;

    // --- A fragments from global: a[kt] holds K = 4*kt + 2*half + {0,1}
    v2f a[16];
    if (active) {
        const float* xrow = X + (size_t)(m0 + mrow) * D + 2 * half;
        #pragma unroll
        for (int kt = 0; kt < 16; ++kt) {
            a[kt][0] = xrow[4 * kt + 0];
            a[kt][1] = xrow[4 * kt + 1];
        }
    }

    // --- layer 1: H = relu(X @ W1 + b1) -> LDS
    if (active) {
        #pragma unroll
        for (int j = 0; j < 4; ++j) {
            const int col = 16 * j + mrow;
            const float bv = b1[col];
            v8f c;
            #pragma unroll
            for (int i = 0; i < 8; ++i) c[i] = bv;
            #pragma unroll
            for (int kt = 0; kt < 16; ++kt)
                c = wmma4(a[kt], ldsB(w1t, KS1, col, 4 * kt + 2 * half), c);
            #pragma unroll
            for (int r = 0; r < 8; ++r)  // C/D layout: row = r + 8*half
                hs[wid][(r + 8 * half) * HSTR + col] = fmaxf(c[r], 0.0f);
        }
    }
    __syncthreads();

    // --- layer 2: Msg = H @ W2 + b2
    if (active) {
        v2f a2[16];
        #pragma unroll
        for (int kt = 0; kt < 16; ++kt)
            a2[kt] = *(const v2f*)(&hs[wid][mrow * HSTR + 4 * kt + 2 * half]);
        #pragma unroll
        for (int j = 0; j < 4; ++j) {
            const int col = 16 * j + mrow;
            const float bv = b2[col];
            v8f c;
            #pragma unroll
            for (int i = 0; i < 8; ++i) c[i] = bv;
            #pragma unroll
            for (int kt = 0; kt < 16; ++kt)
                c = wmma4(a2[kt], ldsB(w2t, KS1, col, 4 * kt + 2 * half), c);
            #pragma unroll
            for (int r = 0; r < 8; ++r)
                Msg[(size_t)(m0 + r + 8 * half) * D + col] = c[r];
        }
    }
}

// ---------------------------------------------------------------------------
// Kernel 2: scatter-add  aggregated[dst] += messages[src] * w;  degree[dst] += w
// 16 threads per edge, one float4 per thread -> coalesced 256B row reads,
// fp32 atomics without return (global_atomic_add_f32), L2-resident.
// ---------------------------------------------------------------------------
__global__ __launch_bounds__(256) void edge_scatter_kernel(
    const float* __restrict__ Msg,
    const int* __restrict__ srcIdx, const int* __restrict__ dstIdx,
    const float* __restrict__ Wgt,
    float* __restrict__ Agg, float* __restrict__ Deg)
{
    const int gid = blockIdx.x * 256 + threadIdx.x;   // E*16 = 20M < 2^31
    const int e = gid >> 4;
    const int p = gid & 15;
    if (e >= N_EDGES) return;

    const int s = srcIdx[e];
    const int d = dstIdx[e];
    const float w = Wgt[e];

    const float4 m = *(const float4*)(Msg + (size_t)s * D + p * 4);
    float* out = Agg + (size_t)d * D + p * 4;
    unsafeAtomicAdd(out + 0, m.x * w);
    unsafeAtomicAdd(out + 1, m.y * w);
    unsafeAtomicAdd(out + 2, m.z * w);
    unsafeAtomicAdd(out + 3, m.w * w);
    if (p == 0) unsafeAtomicAdd(Deg + d, w);
}

// ---------------------------------------------------------------------------
// Kernel 3: fused gate/update MLPs (K=128 and K=64 GEMMs) + gating + LayerNorm
// One wave per 16-row tile; 4 waves per block; K-major weights in dynamic LDS.
// ---------------------------------------------------------------------------
__global__ __launch_bounds__(128) void fused_update_kernel(
    const float* __restrict__ NodeF,
    const float* __restrict__ Agg, const float* __restrict__ Deg,
    const float* __restrict__ Wg,  const float* __restrict__ bg,
    const float* __restrict__ Wu1, const float* __restrict__ bu1,
    const float* __restrict__ Wu2, const float* __restrict__ bu2,
    const float* __restrict__ gamma, const float* __restrict__ beta,
    float* __restrict__ Out)
{
    extern __shared__ float smem[];
    float* wgt  = smem;                    // [64][KS2]  (Wg transposed,  K=128)
    float* wu1t = wgt  + D * KS2;          // [64][KS2]  (Wu1 transposed, K=128)
    float* wu2t = wu1t + D * KS2;          // [64][KS1]  (Wu2 transposed, K=64)
    float* hsb  = wu2t + D * KS1;          // [4][16*HSTR]

    const int tid = threadIdx.x;
    for (int i = tid; i < 2 * D * D; i += 128) {
        int k = i >> 6, n = i & 63;        // Wg/Wu1 are [k][n] row-major, k in [0,128)
        wgt [n * KS2 + k] = Wg[i];
        wu1t[n * KS2 + k] = Wu1[i];
    }
    for (int i = tid; i < D * D; i += 128) {
        int k = i >> 6, n = i & 63;
        wu2t[n * KS1 + k] = Wu2[i];
    }
    __syncthreads();

    const int wid  = tid >> 5;
    const int lane = tid & 31;
    const int half = lane >> 4;
    const int mrow = lane & 15;
    const int tile = blockIdx.x * 4 + wid;
    const bool active = tile < NT;
    const int m0 = tile * 16;
    float* hs = hsb + wid * (16 * HSTR);

    // --- A fragments over K=128: [ agg/max(deg,1) | node_feats ]
    v2f a[32];
    v8f g[4], o[4];
    if (active) {
        const float dg = Deg[m0 + mrow];
        const float rs = 1.0f / fmaxf(dg, 1.0f);
        const float* arow = Agg   + (size_t)(m0 + mrow) * D;
        const float* nrow = NodeF + (size_t)(m0 + mrow) * D;
        #pragma unroll
        for (int kt = 0; kt < 32; ++kt) {
            #pragma unroll
            for (int v = 0; v < 2; ++v) {
                const int kk = 4 * kt + 2 * half + v;
                a[kt][v] = (kk < D) ? arow[kk] * rs : nrow[kk - D];
            }
        }

        // --- gate = sigmoid(comb @ Wg + bg);  u1 = relu(comb @ Wu1 + bu1)
        #pragma unroll
        for (int j = 0; j < 4; ++j) {
            const int col = 16 * j + mrow;
            const float bgv = bg[col], buv = bu1[col];
            v8f cg, cu;
            #pragma unroll
            for (int i = 0; i < 8; ++i) { cg[i] = bgv; cu[i] = buv; }
            #pragma unroll
            for (int kt = 0; kt < 32; ++kt) {
                const int k = 4 * kt + 2 * half;
                cg = wmma4(a[kt], ldsB(wgt,  KS2, col, k), cg);
                cu = wmma4(a[kt], ldsB(wu1t, KS2, col, k), cu);
            }
            #pragma unroll
            for (int r = 0; r < 8; ++r) {
                g[j][r] = 1.0f / (1.0f + expf(-cg[r]));               // sigmoid
                hs[(r + 8 * half) * HSTR + col] = fmaxf(cu[r], 0.0f); // relu
            }
        }
    }
    __syncthreads();

    if (active) {
        // --- u2 = u1 @ Wu2 + bu2 ; out = g*u2 + (1-g)*node
        v2f a2[16];
        #pragma unroll
        for (int kt = 0; kt < 16; ++kt)
            a2[kt] = *(const v2f*)(&hs[mrow * HSTR + 4 * kt + 2 * half]);
        #pragma unroll
        for (int j = 0; j < 4; ++j) {
            const int col = 16 * j + mrow;
            const float bv = bu2[col];
            v8f c;
            #pragma unroll
            for (int i = 0; i < 8; ++i) c[i] = bv;
            #pragma unroll
            for (int kt = 0; kt < 16; ++kt)
                c = wmma4(a2[kt], ldsB(wu2t, KS1, col, 4 * kt + 2 * half), c);
            #pragma unroll
            for (int r = 0; r < 8; ++r) {
                const float nf = NodeF[(size_t)(m0 + r + 8 * half) * D + col];
                o[j][r] = g[j][r] * c[r] + (1.0f - g[j][r]) * nf;
            }
        }

        // --- LayerNorm over D=64 per row. Row (r + 8*half) lives in VGPR r of
        // the 4 column tiles across the 16 lanes of this half-wave.
        float s[8], q[8];
        #pragma unroll
        for (int r = 0; r < 8; ++r) {
            float sv = 0.0f, qv = 0.0f;
            #pragma unroll
            for (int j = 0; j < 4; ++j) { sv += o[j][r]; qv += o[j][r] * o[j][r]; }
            #pragma unroll
            for (int m = 1; m <= 8; m <<= 1) {   // reduce within 16-lane group
                sv += __shfl_xor(sv, m, 32);
                qv += __shfl_xor(qv, m, 32);
            }
            s[r] = sv * (1.0f / D);
            q[r] = qv * (1.0f / D);
        }
        #pragma unroll
        for (int j = 0; j < 4; ++j) {
            const int col = 16 * j + mrow;
            const float gm = gamma[col], bt = beta[col];
            #pragma unroll
            for (int r = 0; r < 8; ++r) {
                const float mean = s[r];
                const float var  = q[r] - mean * mean;
                const float rsd  = rsqrtf(var + LN_EPS);
                Out[(size_t)(m0 + r + 8 * half) * D + col] =
                    (o[j][r] - mean) * rsd * gm + bt;
            }
        }
    }
}

// ---------------------------------------------------------------------------
extern "C" void kernel_launch(void* const* d_in, const int* in_sizes, int n_in,
                              void* d_out, int out_size, void* d_ws, size_t ws_size,
                              hipStream_t stream) {
    const float* node  = (const float*)d_in[0];
    const float* neigh = (const float*)d_in[1];
    const int*   eidx  = (const int*)  d_in[2];   // [2, E]
    const float* ew    = (const float*)d_in[3];   // [E, 1]
    const float* W1  = (const float*)d_in[4];
    const float* b1  = (const float*)d_in[5];
    const float* W2  = (const float*)d_in[6];
    const float* b2  = (const float*)d_in[7];
    const float* Wg  = (const float*)d_in[8];
    const float* bg  = (const float*)d_in[9];
    const float* Wu1 = (const float*)d_in[10];
    const float* bu1 = (const float*)d_in[11];
    const float* Wu2 = (const float*)d_in[12];
    const float* bu2 = (const float*)d_in[13];
    const float* gam = (const float*)d_in[14];
    const float* bet = (const float*)d_in[15];

    float* out = (float*)d_out;                 // final output; also temp for messages
    float* agg = (float*)d_ws;                  // [N, D]
    float* deg = agg + (size_t)N_NODES * D;     // [N]

    // zero the scatter accumulators (capturable stream op)
    hipMemsetAsync(d_ws, 0, ((size_t)N_NODES * D + N_NODES) * sizeof(float), stream);

    // 1) message MLP (messages staged in d_out, overwritten by kernel 3)
    msg_mlp_kernel<<<(NT + 3) / 4, 128, 0, stream>>>(neigh, W1, b1, W2, b2, out);

    // 2) edge gather -> weight -> scatter-add (16 threads per edge)
    edge_scatter_kernel<<<(N_EDGES * 16) / 256, 256, 0, stream>>>(
        out, eidx, eidx + N_EDGES, ew, agg, deg);

    // 3) fused gate/update/LayerNorm (dynamic LDS ~99 KB)
    const int smem3 = (2 * D * KS2 + D * KS1 + 4 * 16 * HSTR) * (int)sizeof(float);
    hipFuncSetAttribute(reinterpret_cast<const void*>(fused_update_kernel),
                        hipFuncAttributeMaxDynamicSharedMemorySize, smem3);
    fused_update_kernel<<<(NT + 3) / 4, 128, smem3, stream>>>(
        node, agg, deg, Wg, bg, Wu1, bu1, Wu2, bu2, gam, bet, out);
}